// MultiHeadGATLayer_12300786336171
// MI455X (gfx1250) — compile-verified
//
#include <hip/hip_runtime.h>
#include <hip/hip_bf16.h>
#include <stdint.h>

#define N_NODES 4096
#define N_EDGES 131072
#define IN_F    256
#define OUT_F   64
#define HEADS   4

typedef __attribute__((ext_vector_type(2))) float v2f;
typedef __attribute__((ext_vector_type(8))) float v8f;

// ---------------- stage 0: zero scratch accumulators ----------------
__global__ void zero_f32(float* __restrict__ p, int n) {
    int t = blockIdx.x * blockDim.x + threadIdx.x;
    if (t < n) p[t] = 0.0f;
}

// ---------------- stage 1: Wh = h @ W^T  (V_WMMA_F32_16X16X4_F32) ----------------
// grid (256 M-tiles, 4 heads), block 128 = 4 waves; wave w covers N-tile w (OUT=64).
__global__ void gemm_wh(const float* __restrict__ h, const float* __restrict__ W,
                        float* __restrict__ Wh) {
    const int mt   = blockIdx.x;           // 0..255
    const int head = blockIdx.y;           // 0..3
    const int wave = threadIdx.x >> 5;     // 0..3  -> N tile
    const int lane = threadIdx.x & 31;
    const int mrow = lane & 15;            // M for A, N for B
    const int kh   = (lane >> 4) * 2;      // lanes 0-15 -> K={0,1}, lanes 16-31 -> K={2,3}

    const float* __restrict__ hA = h + (size_t)(mt * 16 + mrow) * IN_F;
    const float* __restrict__ wB = W + (size_t)(head * OUT_F + wave * 16 + mrow) * IN_F;

    v8f c = {};
#pragma unroll 4
    for (int kb = 0; kb < IN_F; kb += 4) {
        v2f a, b;
        a.x = hA[kb + kh];       // A[m][kh]
        a.y = hA[kb + kh + 1];   // A[m][kh+1]
        b.x = wB[kb + kh];       // B[kh][n]   (B = W^T tile)
        b.y = wB[kb + kh + 1];   // B[kh+1][n]
        c = __builtin_amdgcn_wmma_f32_16x16x4_f32(false, a, false, b,
                                                  (short)0, c, false, false);
    }
    // D layout: VGPR r -> M=r (lanes 0-15) / M=8+r (lanes 16-31), N = lane&15
    const int mbase = (lane >> 4) * 8;
    float* __restrict__ out =
        Wh + ((size_t)head * N_NODES + mt * 16) * OUT_F + wave * 16 + (lane & 15);
#pragma unroll
    for (int r = 0; r < 8; ++r)
        out[(size_t)(mbase + r) * OUT_F] = c[r];
}

// ---------------- stage 2: per-node attention dot products ----------------
__global__ void node_dots(const float* __restrict__ Wh, const float* __restrict__ a,
                          float* __restrict__ p1, float* __restrict__ p2) {
    int t = blockIdx.x * blockDim.x + threadIdx.x;      // HEADS*N
    if (t >= HEADS * N_NODES) return;
    int head = t / N_NODES, n = t % N_NODES;
    const float* wr = Wh + ((size_t)head * N_NODES + n) * OUT_F;
    const float* a1 = a + head * 2 * OUT_F;
    const float* a2 = a1 + OUT_F;
    float s1 = 0.f, s2 = 0.f;
#pragma unroll 8
    for (int o = 0; o < OUT_F; ++o) { float v = wr[o]; s1 += v * a1[o]; s2 += v * a2[o]; }
    p1[t] = s1;
    p2[t] = s2;
}

// ---------------- stage 3: edge scores + row max (int-punned atomicMax) ----------------
__global__ void edge_pass1(const long long* __restrict__ ei,
                           const float* __restrict__ p1, const float* __restrict__ p2,
                           float* __restrict__ ebuf, float* __restrict__ mbuf) {
    int t = blockIdx.x * blockDim.x + threadIdx.x;      // HEADS*E
    if (t >= HEADS * N_EDGES) return;
    int head = t / N_EDGES, k = t % N_EDGES;
    int i = (int)ei[k];
    int j = (int)ei[N_EDGES + k];
    float e = p1[head * N_NODES + i] + p2[head * N_NODES + j];
    e = (e > 0.f) ? e : 0.2f * e;                       // leaky_relu(0.2)
    ebuf[t] = e;
    // m_i = max(0, max_e). max(e,0) >= 0 so int ordering == float ordering.
    atomicMax((int*)&mbuf[head * N_NODES + i], __float_as_int(fmaxf(e, 0.0f)));
}

// ---------------- stage 4: softmax denominator corrections ----------------
__global__ void edge_pass2(const long long* __restrict__ ei,
                           const float* __restrict__ ebuf, const float* __restrict__ mbuf,
                           float* __restrict__ zcorr) {
    int t = blockIdx.x * blockDim.x + threadIdx.x;      // HEADS*E
    if (t >= HEADS * N_EDGES) return;
    int head = t / N_EDGES, k = t % N_EDGES;
    int i = (int)ei[k];
    float m = mbuf[head * N_NODES + i];
    float v = __expf(ebuf[t] - m) - __expf(-m);
    atomicAdd(&zcorr[head * N_NODES + i], v);
}

// ---------------- stage 5: row stats ----------------
__global__ void row_stats(const float* __restrict__ mbuf, const float* __restrict__ zcorr,
                          float* __restrict__ zrecip, float* __restrict__ basealpha) {
    int t = blockIdx.x * blockDim.x + threadIdx.x;      // HEADS*N
    if (t >= HEADS * N_NODES) return;
    float m  = mbuf[t];
    float em = __expf(-m);
    float z  = (float)N_NODES * em + zcorr[t];          // dense-softmax denominator
    float zr = 1.0f / z;
    zrecip[t]    = zr;
    basealpha[t] = em * zr;                             // alpha at all non-edge cells
}

__global__ void row_base(const float* __restrict__ basealpha, float* __restrict__ rowbase) {
    int i = blockIdx.x * blockDim.x + threadIdx.x;      // N
    if (i >= N_NODES) return;
    float s = 0.f;
#pragma unroll
    for (int h = 0; h < HEADS; ++h) s += basealpha[h * N_NODES + i];
    rowbase[i] = s * (1.0f / HEADS);
}

// ---------------- stage 6: dense attn_mean fill (bandwidth bound, ~67MB) ----------------
__global__ void fill_attn(const float* __restrict__ rowbase, float4* __restrict__ attn4) {
    int t = blockIdx.x * blockDim.x + threadIdx.x;      // N*N/4
    if (t >= N_NODES * N_NODES / 4) return;
    int i = t >> 10;                                    // 1024 float4 per row
    float b = rowbase[i];
    attn4[t] = make_float4(b, b, b, b);
}

// ---------------- stage 7: scatter edge attn values (set semantics) ----------------
__global__ void scatter_attn(const long long* __restrict__ ei, const float* __restrict__ ebuf,
                             const float* __restrict__ mbuf, const float* __restrict__ zrecip,
                             float* __restrict__ attn) {
    int k = blockIdx.x * blockDim.x + threadIdx.x;      // E
    if (k >= N_EDGES) return;
    int i = (int)ei[k];
    int j = (int)ei[N_EDGES + k];
    float s = 0.f;
#pragma unroll
    for (int h = 0; h < HEADS; ++h) {
        float m = mbuf[h * N_NODES + i];
        s += __expf(ebuf[h * N_EDGES + k] - m) * zrecip[h * N_NODES + i];
    }
    attn[(size_t)i * N_NODES + j] = s * (1.0f / HEADS);
}

// ---------------- stage 8: column sums of Wh ----------------
__global__ void colsum_k(const float* __restrict__ Wh, float* __restrict__ colsum) {
    int head = blockIdx.x;                              // HEADS
    int o    = threadIdx.x;                             // OUT_F
    const float* base = Wh + (size_t)head * N_NODES * OUT_F + o;
    float s = 0.f;
    for (int n = 0; n < N_NODES; ++n) s += base[(size_t)n * OUT_F];
    colsum[head * OUT_F + o] = s;
}

// ---------------- stage 9: h' baseline: base_i * colsum ----------------
__global__ void hp_init(const float* __restrict__ basealpha, const float* __restrict__ colsum,
                        float* __restrict__ hp) {
    int t = blockIdx.x * blockDim.x + threadIdx.x;      // HEADS*N*OUT
    if (t >= HEADS * N_NODES * OUT_F) return;
    int o    = t & (OUT_F - 1);
    int n    = (t >> 6) & (N_NODES - 1);
    int head = t >> 18;
    hp[t] = basealpha[head * N_NODES + n] * colsum[head * OUT_F + o];
}

// ---------------- stage 10: sparse correction: (alpha_e - base)*Wh[j] ----------------
// 4 (edge,head) pairs per 256-thread block; 64 consecutive lanes -> coalesced atomics.
__global__ void hp_edges(const long long* __restrict__ ei, const float* __restrict__ ebuf,
                         const float* __restrict__ mbuf, const float* __restrict__ zrecip,
                         const float* __restrict__ Wh, float* __restrict__ hp) {
    int pair = blockIdx.x * 4 + (threadIdx.x >> 6);     // k*HEADS + head
    int o    = threadIdx.x & (OUT_F - 1);
    if (pair >= N_EDGES * HEADS) return;
    int head = pair & (HEADS - 1);
    int k    = pair >> 2;
    int i = (int)ei[k];
    int j = (int)ei[N_EDGES + k];
    float m  = mbuf[head * N_NODES + i];
    float w  = (__expf(ebuf[head * N_EDGES + k] - m) - __expf(-m)) * zrecip[head * N_NODES + i];
    atomicAdd(&hp[((size_t)head * N_NODES + i) * OUT_F + o],
              w * Wh[((size_t)head * N_NODES + j) * OUT_F + o]);
}

// ---------------- stage 11: relu + head-concat ----------------
__global__ void out_final(const float* __restrict__ hp, float* __restrict__ out) {
    int t = blockIdx.x * blockDim.x + threadIdx.x;      // N * HEADS*OUT
    if (t >= N_NODES * HEADS * OUT_F) return;
    int i = t >> 8;
    int c = t & 255;
    int head = c >> 6;
    int o    = c & 63;
    float v = hp[((size_t)head * N_NODES + i) * OUT_F + o];
    out[t] = v > 0.f ? v : 0.f;
}

extern "C" void kernel_launch(void* const* d_in, const int* in_sizes, int n_in,
                              void* d_out, int out_size, void* d_ws, size_t ws_size,
                              hipStream_t stream) {
    const float*     h  = (const float*)d_in[0];
    const long long* ei = (const long long*)d_in[1];
    const float*     W  = (const float*)d_in[2];
    const float*     a  = (const float*)d_in[3];

    float* out  = (float*)d_out;                        // N x (HEADS*OUT)
    float* attn = out + (size_t)N_NODES * HEADS * OUT_F; // N x N

    // workspace carve (floats) — ~10.9 MB total
    float* Wh        = (float*)d_ws;                    // H*N*OUT = 1048576
    float* p1        = Wh + 1048576;                    // 16384
    float* p2        = p1 + 16384;                      // 16384
    float* ebuf      = p2 + 16384;                      // H*E = 524288
    float* mbuf      = ebuf + 524288;                   // 16384
    float* zcorr     = mbuf + 16384;                    // 16384 (contiguous w/ mbuf)
    float* zrecip    = zcorr + 16384;                   // 16384
    float* basealpha = zrecip + 16384;                  // 16384
    float* rowbase   = basealpha + 16384;               // 4096
    float* colsum    = rowbase + 4096;                  // 256
    float* hp        = colsum + 256;                    // 1048576

    zero_f32<<<128, 256, 0, stream>>>(mbuf, 2 * HEADS * N_NODES);          // mbuf+zcorr
    gemm_wh<<<dim3(256, HEADS), 128, 0, stream>>>(h, W, Wh);
    node_dots<<<64, 256, 0, stream>>>(Wh, a, p1, p2);
    edge_pass1<<<2048, 256, 0, stream>>>(ei, p1, p2, ebuf, mbuf);
    edge_pass2<<<2048, 256, 0, stream>>>(ei, ebuf, mbuf, zcorr);
    row_stats<<<64, 256, 0, stream>>>(mbuf, zcorr, zrecip, basealpha);
    row_base<<<16, 256, 0, stream>>>(basealpha, rowbase);
    fill_attn<<<16384, 256, 0, stream>>>(rowbase, (float4*)attn);
    scatter_attn<<<512, 256, 0, stream>>>(ei, ebuf, mbuf, zrecip, attn);
    colsum_k<<<HEADS, OUT_F, 0, stream>>>(Wh, colsum);
    hp_init<<<4096, 256, 0, stream>>>(basealpha, colsum, hp);
    hp_edges<<<131072, 256, 0, stream>>>(ei, ebuf, mbuf, zrecip, Wh, hp);
    out_final<<<4096, 256, 0, stream>>>(hp, out);
}